// GlobalReadout_57518202028474
// MI455X (gfx1250) — compile-verified
//
#include <hip/hip_runtime.h>
#include <hip/hip_bf16.h>

#define NODES   500000
#define HID     256
#define NGRAPH  2048
#define NPB     1024    // nodes per block in segment-sum (4 row-subgroups x 256 iters)

typedef __attribute__((ext_vector_type(2))) float v2f;
typedef __attribute__((ext_vector_type(4))) float v4f;
typedef __attribute__((ext_vector_type(8))) float v8f;

// ---------------- zero workspace accumulators ----------------
__global__ void zero_f32_kernel(float* __restrict__ p, int n) {
    int i = blockIdx.x * blockDim.x + threadIdx.x;
    if (i < n) p[i] = 0.0f;
}

// ---------------- segment sum over sorted batch ----------------
// blockDim = 256. Threads are split into 4 subgroups of 64; subgroup `sub`
// processes nodes start+sub, start+sub+4, ... (still sorted), each thread
// handling 4 consecutive features via one b128 non-temporal load (64 threads
// x 16B = one full 1KB node row). Run-length accumulation in registers,
// flushed with float atomics only at graph boundaries. Non-temporal loads
// keep the 512MB h_v stream from evicting the 2MB accumulator out of L2.
__global__ void seg_sum_kernel(const float* __restrict__ hv,
                               const long long* __restrict__ batch,
                               float* __restrict__ sums,      // [NGRAPH][HID]
                               float* __restrict__ counts) {  // [NGRAPH]
    const int t   = threadIdx.x;
    const int sub = t >> 6;            // row subgroup 0..3
    const int fl  = (t & 63) << 2;     // feature base: 0,4,...,252

    long long start = (long long)blockIdx.x * NPB;
    if (start >= NODES) return;
    long long end = start + NPB;
    if (end > NODES) end = NODES;

    long long i = start + sub;
    if (i >= end) return;

    int  cur = (int)batch[i];
    v4f  acc = {0.0f, 0.0f, 0.0f, 0.0f};
    int  run = 0;

    for (; i < end; i += 4) {
        int g = (int)batch[i];
        if (g != cur) {
            float* s = &sums[(long long)cur * HID + fl];
            atomicAdd(s + 0, acc.x);
            atomicAdd(s + 1, acc.y);
            atomicAdd(s + 2, acc.z);
            atomicAdd(s + 3, acc.w);
            if ((t & 63) == 0) atomicAdd(&counts[cur], (float)run);
            acc = (v4f){0.0f, 0.0f, 0.0f, 0.0f};
            run = 0;
            cur = g;
        }
        v4f x = __builtin_nontemporal_load(
            reinterpret_cast<const v4f*>(hv + i * HID + fl));
        acc += x;
        run += 1;
    }
    float* s = &sums[(long long)cur * HID + fl];
    atomicAdd(s + 0, acc.x);
    atomicAdd(s + 1, acc.y);
    atomicAdd(s + 2, acc.z);
    atomicAdd(s + 3, acc.w);
    if ((t & 63) == 0) atomicAdd(&counts[cur], (float)run);
}

// ---------------- mean: sums /= max(count,1) (in place) ----------------
__global__ void mean_kernel(float* __restrict__ sums, const float* __restrict__ counts) {
    int i = blockIdx.x * blockDim.x + threadIdx.x;   // over NGRAPH*HID
    if (i < NGRAPH * HID) {
        float c = counts[i / HID];
        sums[i] = sums[i] / fmaxf(c, 1.0f);
    }
}

// ---------------- 256x256 transpose (W[k][n] -> Wt[n][k]) ----------------
__global__ void transpose256_kernel(const float* __restrict__ W, float* __restrict__ Wt) {
    __shared__ float tile[16][17];
    int bx = blockIdx.x * 16, by = blockIdx.y * 16;
    tile[threadIdx.y][threadIdx.x] = W[(by + threadIdx.y) * HID + (bx + threadIdx.x)];
    __syncthreads();
    Wt[(bx + threadIdx.y) * HID + (by + threadIdx.x)] = tile[threadIdx.x][threadIdx.y];
}

// ---------------- GEMM + bias + ReLU via V_WMMA_F32_16X16X4_F32 ----------------
// C[M,N] = relu(A[M,K] @ B[K,N] + bias[N]), with B supplied transposed (Bt[N][K]).
// One wave (32 threads) per 16x16 output tile. EXEC is all-1s (no divergence).
//
// Fragment layouts per CDNA5 ISA (32-bit, wave32):
//   A 16x4 : lane l (l15 = l&15, half = l>>4): M = m0+l15, VGPR0/1 = K = k+2*half, k+2*half+1
//   B 4x16 : lane l: N = n0+l15,               VGPR0/1 = K = k+2*half, k+2*half+1
//   C 16x16: VGPR v: M = m0 + v + 8*half, N = n0 + l15
__global__ void gemm_wmma_kernel(const float* __restrict__ A,
                                 const float* __restrict__ Bt,
                                 const float* __restrict__ bias,
                                 float* __restrict__ C,
                                 int M, int N, int K) {
    const int lane = threadIdx.x;
    const int half = lane >> 4;
    const int l15  = lane & 15;
    const int m0 = blockIdx.x * 16;
    const int n0 = blockIdx.y * 16;

    const float* arow = A  + (size_t)(m0 + l15) * K + 2 * half;
    const float* brow = Bt + (size_t)(n0 + l15) * K + 2 * half;

    v8f acc = {};
    for (int k = 0; k < K; k += 4) {
        v2f a = *reinterpret_cast<const v2f*>(arow + k);
        v2f b = *reinterpret_cast<const v2f*>(brow + k);
        acc = __builtin_amdgcn_wmma_f32_16x16x4_f32(
            /*neg_a=*/false, a, /*neg_b=*/false, b,
            /*c_mod=*/(short)0, acc, /*reuse_a=*/false, /*reuse_b=*/false);
    }

    const float bval = bias[n0 + l15];
#pragma unroll
    for (int v = 0; v < 8; ++v) {
        int row = m0 + v + 8 * half;
        float r = acc[v] + bval;
        C[(size_t)row * N + (n0 + l15)] = r > 0.0f ? r : 0.0f;
    }
}

// ---------------- final 256 -> 1 layer + empty-graph mask ----------------
__global__ void final_layer_kernel(const float* __restrict__ x,      // [NGRAPH][HID]
                                   const float* __restrict__ W3,     // [HID]
                                   const float* __restrict__ b3,     // [1]
                                   const float* __restrict__ counts, // [NGRAPH]
                                   float* __restrict__ out) {        // [NGRAPH]
    __shared__ float red[256];
    const int g = blockIdx.x, t = threadIdx.x;
    red[t] = x[(size_t)g * HID + t] * W3[t];
    __syncthreads();
    for (int s = 128; s > 0; s >>= 1) {
        if (t < s) red[t] += red[t + s];
        __syncthreads();
    }
    if (t == 0) out[g] = (counts[g] > 0.0f) ? (red[0] + b3[0]) : 0.0f;
}

extern "C" void kernel_launch(void* const* d_in, const int* in_sizes, int n_in,
                              void* d_out, int out_size, void* d_ws, size_t ws_size,
                              hipStream_t stream) {
    const float*     h_v   = (const float*)d_in[0];
    /* d_in[1] = edge_index (unused by reference) */
    const long long* batch = (const long long*)d_in[2];
    const float*     W1    = (const float*)d_in[3];
    const float*     b1    = (const float*)d_in[4];
    const float*     W2    = (const float*)d_in[5];
    const float*     b2    = (const float*)d_in[6];
    const float*     W3    = (const float*)d_in[7];
    const float*     b3    = (const float*)d_in[8];
    float* out = (float*)d_out;

    // workspace layout (bytes)
    char* base = (char*)d_ws;
    float* sums   = (float*)(base);                                   // 2048*256 f32 (becomes pooled)
    float* counts = (float*)(base + (size_t)NGRAPH * HID * 4);        // 2048 f32
    float* W1t    = (float*)(base + (size_t)NGRAPH * HID * 4 + NGRAPH * 4);
    float* W2t    = W1t + HID * HID;
    float* x1     = W2t + HID * HID;                                  // 2048*256 f32
    float* x2     = x1 + (size_t)NGRAPH * HID;                        // 2048*256 f32

    // 1) zero accumulators
    {
        int n = NGRAPH * HID + NGRAPH;
        zero_f32_kernel<<<(n + 255) / 256, 256, 0, stream>>>(sums, n);
    }
    // 2) segment sum + counts (sorted batch -> run-length flush, b128 NT loads)
    {
        int blocks = (NODES + NPB - 1) / NPB;
        seg_sum_kernel<<<blocks, 256, 0, stream>>>(h_v, batch, sums, counts);
    }
    // 3) mean pooling (in place)
    {
        int n = NGRAPH * HID;
        mean_kernel<<<(n + 255) / 256, 256, 0, stream>>>(sums, counts);
    }
    // 4) transpose weights so WMMA B-fragments read contiguous K
    {
        dim3 blk(16, 16), grd(HID / 16, HID / 16);
        transpose256_kernel<<<grd, blk, 0, stream>>>(W1, W1t);
        transpose256_kernel<<<grd, blk, 0, stream>>>(W2, W2t);
    }
    // 5) MLP layers 1 & 2 via WMMA f32 (exact precision), one wave per tile
    {
        dim3 grd(NGRAPH / 16, HID / 16);
        gemm_wmma_kernel<<<grd, 32, 0, stream>>>(sums, W1t, b1, x1, NGRAPH, HID, HID);
        gemm_wmma_kernel<<<grd, 32, 0, stream>>>(x1,   W2t, b2, x2, NGRAPH, HID, HID);
    }
    // 6) final projection + mask
    final_layer_kernel<<<NGRAPH, HID, 0, stream>>>(x2, W3, b3, counts, out);
}